// WTA_4629974745675
// MI455X (gfx1250) — compile-verified
//
#include <hip/hip_runtime.h>
#include <hip/hip_bf16.h>
#include <stdint.h>

// Problem constants (reference: x[128,256,32,32] f32, topk=5)
#define BLOCK   1024
#define NPB     (256*32*32)            // 262144 elements per batch row
#define VPT     (NPB / 4 / BLOCK)      // 64 float4 chunks per thread
#define KMAX    5
#define CAP     64                     // per-thread candidate buffer (scratch)
#define TIECAP  64

typedef int v4i __attribute__((ext_vector_type(4)));
typedef __attribute__((address_space(1))) v4i* g4p;   // global int32x4*
typedef __attribute__((address_space(3))) v4i* l4p;   // LDS    int32x4*
typedef __attribute__((address_space(3))) void AS3v;

// ---- CDNA5 async LDS->global store (ASYNCcnt path) ------------------------
__device__ __forceinline__ void store_zero_async_b128(float* gdst, float* lsrc) {
#if defined(__AMDGCN__)
#if __has_builtin(__builtin_amdgcn_global_store_async_from_lds_b128)
  __builtin_amdgcn_global_store_async_from_lds_b128((g4p)(v4i*)gdst, (l4p)(v4i*)lsrc, 0, 0);
#else
  unsigned long long ga = (unsigned long long)(uintptr_t)gdst;
  unsigned int       la = (unsigned int)(uintptr_t)(AS3v*)lsrc;
  asm volatile("global_store_async_from_lds_b128 %0, %1, off"
               :: "v"(ga), "v"(la) : "memory");
#endif
#else
  (void)gdst; (void)lsrc;
#endif
}

__device__ __forceinline__ void wait_async0() {
#if defined(__AMDGCN__)
#if __has_builtin(__builtin_amdgcn_s_wait_asynccnt)
  __builtin_amdgcn_s_wait_asynccnt(0);
#else
  asm volatile("s_wait_asynccnt 0x0" ::: "memory");
#endif
#endif
}

// ---------------------------------------------------------------------------
__global__ __launch_bounds__(BLOCK)
void topk_mask_kernel(const float* __restrict__ x,
                      const int*   __restrict__ kp,
                      float*       __restrict__ out)
{
  __shared__ float s_vals[BLOCK * KMAX];   // 20 KB: per-thread sorted top-5 values
  __shared__ float s_zero[BLOCK * 4];      // 16 KB: zero staging for async stores
  __shared__ int   s_tie[TIECAP];
  __shared__ int   s_tiecnt;

  const int t = threadIdx.x;
  const int b = blockIdx.x;

  // init zero staging + tie list (visible via the reduction barriers below)
  ((float4*)s_zero)[t] = make_float4(0.f, 0.f, 0.f, 0.f);
  if (t == 0) s_tiecnt = 0;

  int k = kp[0];
  k = (k < 1) ? 1 : ((k > KMAX) ? KMAX : k);

  const float4* xb = (const float4*)(x + (size_t)b * NPB);

  const float NEG_INF = -__builtin_inff();
  float v0 = NEG_INF, v1 = NEG_INF, v2 = NEG_INF, v3 = NEG_INF, v4 = NEG_INF;

  // per-thread candidate buffer (lands in scratch; ~25 entries expected)
  float bval[CAP];
  int   bidx[CAP];
  int   cnt = 0;

  auto proc = [&](float nv, int ni) {
    // record BEFORE updating: running v4 <= global k-th value, so every
    // element >= global threshold is guaranteed to be recorded.
    if (nv >= v4) {
      if (cnt < CAP) { bval[cnt] = nv; bidx[cnt] = ni; ++cnt; }
    }
    // branchless sorted-insert (values only): 9 min/max ops
    float a;
    a = fmaxf(v0, nv); nv = fminf(v0, nv); v0 = a;
    a = fmaxf(v1, nv); nv = fminf(v1, nv); v1 = a;
    a = fmaxf(v2, nv); nv = fminf(v2, nv); v2 = a;
    a = fmaxf(v3, nv); nv = fminf(v3, nv); v3 = a;
    a = fmaxf(v4, nv);                     v4 = a;
  };

  // ---- Phase A: coalesced streaming scan (memory-bound) -------------------
  #pragma unroll 4
  for (int i = 0; i < VPT; ++i) {
    int j = i * BLOCK + t;
    float4 d = xb[j];
    int e = j * 4;
    proc(d.x, e + 0);
    proc(d.y, e + 1);
    proc(d.z, e + 2);
    proc(d.w, e + 3);
  }

  // ---- Phase B: block tree-reduction of sorted 5-value lists --------------
  s_vals[t*KMAX+0] = v0; s_vals[t*KMAX+1] = v1; s_vals[t*KMAX+2] = v2;
  s_vals[t*KMAX+3] = v3; s_vals[t*KMAX+4] = v4;
  __syncthreads();

  for (int stride = BLOCK / 2; stride > 0; stride >>= 1) {
    if (t < stride) {
      float* a  = &s_vals[(size_t)t * KMAX];
      float* bb = &s_vals[(size_t)(t + stride) * KMAX];
      float m[KMAX];
      int ia = 0, ib = 0;
      #pragma unroll
      for (int o = 0; o < KMAX; ++o) {
        float av = a[ia], bv = bb[ib];
        bool ta = (av >= bv);
        m[o] = ta ? av : bv;
        ia += ta ? 1 : 0;
        ib += ta ? 0 : 1;
      }
      #pragma unroll
      for (int o = 0; o < KMAX; ++o) a[o] = m[o];
    }
    __syncthreads();
  }

  float g0 = s_vals[0], g1 = s_vals[1], g2 = s_vals[2], g3 = s_vals[3], g4 = s_vals[4];
  float T = (k == 1) ? g0 : (k == 2) ? g1 : (k == 3) ? g2 : (k == 4) ? g3 : g4;
  int   c = (g0 > T) + (g1 > T) + (g2 > T) + (g3 > T) + (g4 > T);
  int   need = k - c;   // >= 1; number of tie (==T) elements to mark, lowest index first

  // ---- Phase C: stream zeros via async LDS->global DMA --------------------
  float* ob = out + (size_t)b * NPB;
  for (int i = 0; i < VPT; ++i) {
    int j = i * BLOCK + t;
    store_zero_async_b128(ob + (size_t)j * 4, &s_zero[t * 4]);
  }

  // overlap: collect exact-tie candidates while async stores are in flight
  for (int q = 0; q < cnt; ++q) {
    if (bval[q] == T) {
      int p = atomicAdd(&s_tiecnt, 1);
      if (p < TIECAP) s_tie[p] = bidx[q];
    }
  }

  wait_async0();        // this wave's zero stores done
  __syncthreads();      // all waves' zero stores done -> safe to write ones

  // definite ones: exactly the c elements strictly greater than T
  for (int q = 0; q < cnt; ++q)
    if (bval[q] > T) ob[bidx[q]] = 1.0f;

  // tie ones: k-c smallest indices among elements == T (matches lax.top_k)
  if (t == 0) {
    int m = s_tiecnt; if (m > TIECAP) m = TIECAP;
    for (int r = 0; r < need; ++r) {
      int best = 0x7fffffff, bp = -1;
      for (int q = 0; q < m; ++q) {
        int ix = s_tie[q];
        if (ix < best) { best = ix; bp = q; }
      }
      if (bp >= 0) { ob[best] = 1.0f; s_tie[bp] = 0x7fffffff; }
    }
  }
}

// ---------------------------------------------------------------------------
extern "C" void kernel_launch(void* const* d_in, const int* in_sizes, int n_in,
                              void* d_out, int out_size, void* d_ws, size_t ws_size,
                              hipStream_t stream) {
  (void)n_in; (void)out_size; (void)d_ws; (void)ws_size;
  const float* x  = (const float*)d_in[0];
  const int*   kp = (const int*)d_in[1];
  float*       out = (float*)d_out;
  int B = in_sizes[0] / NPB;   // 128 batch rows
  topk_mask_kernel<<<B, BLOCK, 0, stream>>>(x, kp, out);
}